// SparseRepPointsHead_60979945669322
// MI455X (gfx1250) — compile-verified
//
#include <hip/hip_runtime.h>
#include <math.h>

// ---------------- problem constants (fixed by setup_inputs) ----------------
#define BB     4        // batch
#define CC     256      // channels
#define KK     100      // top-k
#define PP     9        // num points
#define NOBJ   300      // 3 scales * K
#define MROWS  (BB*NOBJ)   // 1200
#define KPATCH (CC*9)      // 2304
#define FIN    (2*PP*CC)   // 4608
#define NCLS   80

// d_out layout (floats): cls(1,4,300,80)=96000 | box(1,4,300,4)=4800 | objs0 60800 | objs1 15200 | objs2 3800
#define CLS_OFF   0
#define BOX_OFF   96000
#define OBJ0_OFF  100800
#define OBJ1_OFF  161600
#define OBJ2_OFF  176800

// workspace layout (float offsets, all even -> 8B aligned)
#define WS_OBJRAW   0          // 79800 (scale sub-offsets 0 / 60800 / 76000)
#define WS_STATS    79808      // 6
#define WS_TOPKI    79816      // 1200 ints
#define WS_XYS      81016      // 2400
#define WS_SCRATCH  83416      // 12*15200
#define WS_APATCH   265816     // 1200*2304
#define WS_Y        3030616    // 1200*256
#define WS_Z        3337816    // 1200*18
#define WS_X        3359416    // 1200*4608
#define WS_H1       8889016    // 1200*256
#define WS_H2       9196216
#define WS_BH1      9503416
#define WS_BH2      9810616
#define WS_BOXRAW   10117816   // 4800

typedef __attribute__((ext_vector_type(2))) float v2f;
typedef __attribute__((ext_vector_type(8))) float v8f;

__device__ __forceinline__ float sigm(float v) { return 1.f / (1.f + __expf(-v)); }

// ------------- objectness 3x3 conv (1 out channel), raw output -------------
__global__ void __launch_bounds__(256) obj_conv_kernel(
    const float* __restrict__ feat, const float* __restrict__ w,
    const float* __restrict__ obj_b, int s, int H, int W,
    float* __restrict__ objraw)
{
    int n = BB * H * W;
    int i = blockIdx.x * blockDim.x + threadIdx.x;
    if (i >= n) return;
    int b = i / (H * W);
    int rem = i - b * H * W;
    int y = rem / W, x = rem - y * W;
    float acc = obj_b[s];
    const float* fb = feat + (size_t)b * CC * H * W;
    for (int cin = 0; cin < CC; ++cin) {
        const float* fp = fb + (size_t)cin * H * W;
        const float* wp = w + cin * 9;
        #pragma unroll
        for (int dy = -1; dy <= 1; ++dy) {
            int yy = y + dy;
            if ((unsigned)yy >= (unsigned)H) continue;
            const float* frow = fp + (size_t)yy * W;
            #pragma unroll
            for (int dx = -1; dx <= 1; ++dx) {
                int xx = x + dx;
                if ((unsigned)xx >= (unsigned)W) continue;
                acc = fmaf(frow[xx], wp[(dy + 1) * 3 + (dx + 1)], acc);
            }
        }
    }
    objraw[i] = acc;
}

// ------------- deterministic batch stats (sum, sumsq) per scale -------------
__global__ void __launch_bounds__(256) reduce_stats_kernel(
    const float* __restrict__ objraw, float* __restrict__ stats)
{
    constexpr int OFF[3] = {0, 60800, 76000};
    constexpr int NN[3]  = {60800, 15200, 3800};
    int s = blockIdx.x;
    const float* p = objraw + OFF[s];
    int n = NN[s];
    float s1 = 0.f, s2 = 0.f;
    for (int i = threadIdx.x; i < n; i += blockDim.x) { float v = p[i]; s1 += v; s2 += v * v; }
    __shared__ float sa[256], sb[256];
    sa[threadIdx.x] = s1; sb[threadIdx.x] = s2;
    __syncthreads();
    for (int off = 128; off > 0; off >>= 1) {
        if (threadIdx.x < off) { sa[threadIdx.x] += sa[threadIdx.x + off]; sb[threadIdx.x] += sb[threadIdx.x + off]; }
        __syncthreads();
    }
    if (threadIdx.x == 0) { stats[2 * s] = sa[0]; stats[2 * s + 1] = sb[0]; }
}

// ------------- batchnorm + sigmoid -> objs output region -------------
__global__ void __launch_bounds__(256) obj_finalize_kernel(
    const float* __restrict__ objraw, const float* __restrict__ stats,
    const float* __restrict__ gamma, const float* __restrict__ beta,
    int s, int n, float* __restrict__ outp)
{
    int i = blockIdx.x * blockDim.x + threadIdx.x;
    if (i >= n) return;
    float mean = stats[2 * s] / (float)n;
    float var  = stats[2 * s + 1] / (float)n - mean * mean;
    float v = (objraw[i] - mean) * rsqrtf(var + 1e-5f) * gamma[s] + beta[s];
    outp[i] = sigm(v);
}

// ------------- per (b,scale) iterative top-K (block = 256 threads) -------------
__global__ void __launch_bounds__(256) topk_kernel(
    const float* __restrict__ out_base, float* __restrict__ scratch,
    int* __restrict__ topki, float* __restrict__ xys)
{
    constexpr int SH[3] = {100, 50, 25}, SW[3] = {152, 76, 38};
    constexpr int OOFF[3] = {OBJ0_OFF, OBJ1_OFF, OBJ2_OFF};
    int blk = blockIdx.x;            // b*3 + s
    int b = blk / 3, s = blk % 3;
    int H = SH[s], W = SW[s], n = H * W;
    const float* p = out_base + OOFF[s] + (size_t)b * n;
    float* sc = scratch + (size_t)blk * 15200;
    for (int i = threadIdx.x; i < n; i += blockDim.x) sc[i] = p[i];
    __syncthreads();
    __shared__ float sval[256];
    __shared__ int   sidx[256];
    for (int r = 0; r < KK; ++r) {
        float best = -3.4e38f; int bi = 0x7fffffff;
        for (int i = threadIdx.x; i < n; i += blockDim.x) {
            float v = sc[i];
            if (v > best || (v == best && i < bi)) { best = v; bi = i; }
        }
        sval[threadIdx.x] = best; sidx[threadIdx.x] = bi;
        __syncthreads();
        for (int off = 128; off > 0; off >>= 1) {
            if (threadIdx.x < off) {
                float v2 = sval[threadIdx.x + off]; int i2 = sidx[threadIdx.x + off];
                if (v2 > sval[threadIdx.x] || (v2 == sval[threadIdx.x] && i2 < sidx[threadIdx.x])) {
                    sval[threadIdx.x] = v2; sidx[threadIdx.x] = i2;
                }
            }
            __syncthreads();
        }
        if (threadIdx.x == 0) {
            int idx = sidx[0];
            topki[blk * KK + r] = idx;
            int row = idx / W, col = idx % W;
            int o = s * KK + r;
            xys[((size_t)b * NOBJ + o) * 2 + 0] = (float)row / (float)H;
            xys[((size_t)b * NOBJ + o) * 2 + 1] = (float)col / (float)W;
            sc[idx] = -3.4e38f;
        }
        __syncthreads();
    }
}

// ------------- gather 3x3x256 zero-padded input patches at top-k positions -------------
__global__ void __launch_bounds__(256) patch_gather_kernel(
    const float* __restrict__ f0, const float* __restrict__ f1, const float* __restrict__ f2,
    const int* __restrict__ topki, float* __restrict__ A)
{
    constexpr int SH[3] = {100, 50, 25}, SW[3] = {152, 76, 38};
    int r = blockIdx.x;               // 0..1199 = b*300 + o
    int b = r / NOBJ, o = r % NOBJ, s = o / KK, k = o % KK;
    int H = SH[s], W = SW[s];
    const float* f = (s == 0) ? f0 : ((s == 1) ? f1 : f2);
    int idx = topki[(b * 3 + s) * KK + k];
    int y = idx / W, x = idx % W;
    const float* fb = f + (size_t)b * CC * H * W;
    float* arow = A + (size_t)r * KPATCH;
    for (int t = threadIdx.x; t < KPATCH; t += blockDim.x) {
        int cin = t / 9, j = t % 9, dy = j / 3 - 1, dx = j % 3 - 1;
        int yy = y + dy, xx = x + dx;
        float v = 0.f;
        if ((unsigned)yy < (unsigned)H && (unsigned)xx < (unsigned)W)
            v = fb[((size_t)cin * H + yy) * W + xx];
        arow[t] = v;
    }
}

// ------------- WMMA fp32 GEMM: D = act(A(MxK) @ W(NxK)^T + bias), row-major -------------
// one wave per 16x16 tile; V_WMMA_F32_16X16X4_F32; both operands contiguous float2 loads.
__global__ void __launch_bounds__(256) gemm_xwt_kernel(
    const float* __restrict__ A, const float* __restrict__ Wm,
    const float* __restrict__ bias, float* __restrict__ D,
    int M, int N, int Kd, int act)
{
    int wave = threadIdx.x >> 5;
    int lane = threadIdx.x & 31;
    int tilesN = (N + 15) >> 4;
    int tileId = blockIdx.x * 8 + wave;
    int tm = tileId / tilesN;
    int tn = tileId - tm * tilesN;
    if (tm * 16 >= M) return;         // wave-uniform: EXEC stays all-1s for WMMA
    int half = lane >> 4;             // lanes 16-31 hold K+2,K+3 operand slices
    int l16  = lane & 15;
    int row  = tm * 16 + l16;         // A row (M multiple of 16 -> always valid)
    int col  = tn * 16 + l16;         // output column == weight row
    int colc = (col < N) ? col : (N - 1);
    const float* arow = A  + (size_t)row  * Kd + 2 * half;
    const float* wrow = Wm + (size_t)colc * Kd + 2 * half;
    v8f acc = {};
    #pragma unroll 8
    for (int k = 0; k < Kd; k += 4) {
        v2f av = *(const v2f*)(arow + k);   // A[row][k+2h .. k+2h+1]
        v2f bv = *(const v2f*)(wrow + k);   // B[k+2h .. k+2h+1][col] = W[col][k+2h ..]
        acc = __builtin_amdgcn_wmma_f32_16x16x4_f32(false, av, false, bv,
                                                    (short)0, acc, false, false);
    }
    float bval = 0.f;
    if (bias != nullptr && col < N) bval = bias[col];
    #pragma unroll
    for (int i = 0; i < 8; ++i) {
        int rr = tm * 16 + i + 8 * half;    // C/D layout: VGPRi -> M=i (lo half) / i+8 (hi half)
        if (rr < M && col < N) {
            float v = acc[i] + bval;
            if (act == 1)      v = fmaxf(v, 0.f);
            else if (act == 2) v = sigm(v);
            D[(size_t)rr * N + col] = v;
        }
    }
}

// ------------- assemble X = [bilinear-border samples | sine pos-encoding] -------------
__global__ void __launch_bounds__(256) assemble_x_kernel(
    const float* __restrict__ f0, const float* __restrict__ f1, const float* __restrict__ f2,
    const float* __restrict__ Z, const float* __restrict__ xys, float* __restrict__ X)
{
    constexpr int SH[3] = {100, 50, 25}, SW[3] = {152, 76, 38};
    int r = blockIdx.x;
    int b = r / NOBJ, o = r % NOBJ, s = o / KK;
    int H = SH[s], W = SW[s];
    const float* f = (s == 0) ? f0 : ((s == 1) ? f1 : f2);
    const float* fb = f + (size_t)b * CC * H * W;
    float xy0 = xys[r * 2 + 0], xy1 = xys[r * 2 + 1];
    float* xrow = X + (size_t)r * FIN;
    // sampled point features: xrow[p*256 + c]
    for (int t = threadIdx.x; t < KPATCH; t += blockDim.x) {
        int p = t / CC, c = t % CC;
        float gx = 2.f * (Z[(size_t)r * 18 + p * 2 + 0] + xy0) - 1.f;  // indexes W
        float gy = 2.f * (Z[(size_t)r * 18 + p * 2 + 1] + xy1) - 1.f;  // indexes H
        float ix = fminf(fmaxf(((gx + 1.f) * W - 1.f) * 0.5f, 0.f), (float)(W - 1));
        float iy = fminf(fmaxf(((gy + 1.f) * H - 1.f) * 0.5f, 0.f), (float)(H - 1));
        float x0f = floorf(ix), y0f = floorf(iy);
        float wx = ix - x0f, wy = iy - y0f;
        int x0 = min(max((int)x0f, 0), W - 1), x1 = min(max((int)x0f + 1, 0), W - 1);
        int y0 = min(max((int)y0f, 0), H - 1), y1 = min(max((int)y0f + 1, 0), H - 1);
        const float* fc = fb + (size_t)c * H * W;
        float v00 = fc[y0 * W + x0], v01 = fc[y0 * W + x1];
        float v10 = fc[y1 * W + x0], v11 = fc[y1 * W + x1];
        xrow[t] = (v00 * (1.f - wx) + v01 * wx) * (1.f - wy)
                + (v10 * (1.f - wx) + v11 * wx) * wy;
    }
    // DETR sine pos encoding (H=100 rows, W=9 cols, 256 ch): xrow[2304 + p*256 + c]
    int h = o % KK;
    for (int t = threadIdx.x; t < KPATCH; t += blockDim.x) {
        int p = t / CC, c = t % CC;
        float val; int cc;
        if (c < 128) { val = (float)(h + 1) / (100.f + 1e-6f) * 6.283185307179586f; cc = c; }
        else         { val = (float)(p + 1) / (9.f   + 1e-6f) * 6.283185307179586f; cc = c - 128; }
        float e = (float)(2 * (cc >> 1)) / 128.f;
        float tv = val / powf(10000.f, e);
        xrow[KPATCH + t] = (cc & 1) ? cosf(tv) : sinf(tv);
    }
}

// ------------- box = sigmoid-out with first two coords shifted by init_xys -------------
__global__ void __launch_bounds__(256) box_finalize_kernel(
    const float* __restrict__ boxraw, const float* __restrict__ xys, float* __restrict__ out)
{
    int i = blockIdx.x * blockDim.x + threadIdx.x;
    if (i >= MROWS * 4) return;
    int r = i / 4, j = i % 4;
    float v = boxraw[i];
    if (j < 2) v += xys[r * 2 + j];
    out[i] = v;
}

static void launch_gemm(const float* A, const float* Wm, const float* bias, float* D,
                        int M, int N, int Kd, int act, hipStream_t stream)
{
    int tilesN = (N + 15) / 16, tilesM = (M + 15) / 16;
    int blocks = (tilesM * tilesN + 7) / 8;
    gemm_xwt_kernel<<<blocks, 256, 0, stream>>>(A, Wm, bias, D, M, N, Kd, act);
}

extern "C" void kernel_launch(void* const* d_in, const int* in_sizes, int n_in,
                              void* d_out, int out_size, void* d_ws, size_t ws_size,
                              hipStream_t stream)
{
    const float* feat[3]   = {(const float*)d_in[0], (const float*)d_in[1], (const float*)d_in[2]};
    const float* obj_w     = (const float*)d_in[3];   // (3,1,256,3,3)
    const float* obj_b     = (const float*)d_in[4];
    const float* obj_gamma = (const float*)d_in[5];
    const float* obj_beta  = (const float*)d_in[6];
    const float* off_w     = (const float*)d_in[7];   // (256,256,3,3) -> (256,2304)
    const float* off_b     = (const float*)d_in[8];
    const float* offout_w  = (const float*)d_in[9];   // (18,256)
    const float* offout_b  = (const float*)d_in[10];
    const float* fc1_w     = (const float*)d_in[11];  // (256,4608)
    const float* fc2_w     = (const float*)d_in[12];
    const float* cls_w     = (const float*)d_in[13];
    const float* cls_b     = (const float*)d_in[14];
    const float* m1_w      = (const float*)d_in[15];
    const float* m1_b      = (const float*)d_in[16];
    const float* m2_w      = (const float*)d_in[17];
    const float* m2_b      = (const float*)d_in[18];
    const float* m3_w      = (const float*)d_in[19];
    const float* m3_b      = (const float*)d_in[20];

    float* ws = (float*)d_ws;
    float* objraw  = ws + WS_OBJRAW;
    float* stats   = ws + WS_STATS;
    int*   topki   = (int*)(ws + WS_TOPKI);
    float* xys     = ws + WS_XYS;
    float* scratch = ws + WS_SCRATCH;
    float* Apatch  = ws + WS_APATCH;
    float* Ybuf    = ws + WS_Y;
    float* Zbuf    = ws + WS_Z;
    float* Xbuf    = ws + WS_X;
    float* H1      = ws + WS_H1;
    float* H2      = ws + WS_H2;
    float* BH1     = ws + WS_BH1;
    float* BH2     = ws + WS_BH2;
    float* BOXRAW  = ws + WS_BOXRAW;
    float* out     = (float*)d_out;

    const int SH[3] = {100, 50, 25}, SW[3] = {152, 76, 38};
    const int OWS[3]  = {0, 60800, 76000};
    const int OOUT[3] = {OBJ0_OFF, OBJ1_OFF, OBJ2_OFF};

    // objectness: conv -> stats -> norm+sigmoid (written straight into d_out)
    for (int s = 0; s < 3; ++s) {
        int n = BB * SH[s] * SW[s];
        obj_conv_kernel<<<(n + 255) / 256, 256, 0, stream>>>(
            feat[s], obj_w + s * 2304, obj_b, s, SH[s], SW[s], objraw + OWS[s]);
    }
    reduce_stats_kernel<<<3, 256, 0, stream>>>(objraw, stats);
    for (int s = 0; s < 3; ++s) {
        int n = BB * SH[s] * SW[s];
        obj_finalize_kernel<<<(n + 255) / 256, 256, 0, stream>>>(
            objraw + OWS[s], stats, obj_gamma, obj_beta, s, n, out + OOUT[s]);
    }

    // top-K per (b, scale), then per-point offset branch as gathered patch GEMM
    topk_kernel<<<BB * 3, 256, 0, stream>>>(out, scratch, topki, xys);
    patch_gather_kernel<<<MROWS, 256, 0, stream>>>(feat[0], feat[1], feat[2], topki, Apatch);
    launch_gemm(Apatch, off_w,    off_b,    Ybuf, MROWS, 256, KPATCH, 1, stream);  // relu conv3x3@points
    launch_gemm(Ybuf,   offout_w, offout_b, Zbuf, MROWS, 18,  256,    2, stream);  // sigmoid conv1x1

    // sample features at predicted points + pos encoding -> X (1200 x 4608)
    assemble_x_kernel<<<MROWS, 256, 0, stream>>>(feat[0], feat[1], feat[2], Zbuf, xys, Xbuf);

    // FC head (all WMMA fp32)
    launch_gemm(Xbuf, fc1_w, nullptr, H1,            MROWS, 256,  FIN, 1, stream);
    launch_gemm(H1,   fc2_w, nullptr, H2,            MROWS, 256,  256, 1, stream);
    launch_gemm(H2,   cls_w, cls_b,   out + CLS_OFF, MROWS, NCLS, 256, 0, stream);
    launch_gemm(H2,   m1_w,  m1_b,    BH1,           MROWS, 256,  256, 1, stream);
    launch_gemm(BH1,  m2_w,  m2_b,    BH2,           MROWS, 256,  256, 1, stream);
    launch_gemm(BH2,  m3_w,  m3_b,    BOXRAW,        MROWS, 4,    256, 2, stream);
    box_finalize_kernel<<<(MROWS * 4 + 255) / 256, 256, 0, stream>>>(BOXRAW, xys, out + BOX_OFF);
}